// ProposalLayer_80848464380135
// MI455X (gfx1250) — compile-verified
//
#include <hip/hip_runtime.h>
#include <hip/hip_bf16.h>
#include <math.h>

// ProposalLayer for MI455X (gfx1250, wave32).
// Pipeline: decode+key -> single-WGP bitonic sort (128KB LDS) ->
//           upper-triangle IoU suppression bitmask (WMMA outer-sum + ballot,
//           async global->LDS tile staging) ->
//           single-WG sequential greedy scan -> compact top-2000 output.

#define N_ANCH   16384
#define NWORDS   512          // 16384 / 32 column-words per row
#define PROPS    2000
#define NMS_THR  0.7f

typedef __attribute__((ext_vector_type(2))) float v2f;
typedef __attribute__((ext_vector_type(8))) float v8f;

#if __has_builtin(__builtin_amdgcn_global_load_async_to_lds_b128) && \
    __has_builtin(__builtin_amdgcn_global_load_async_to_lds_b32) &&  \
    __has_builtin(__builtin_amdgcn_s_wait_asynccnt)
#define HAS_ASYNC_LDS 1
typedef int v4i_ __attribute__((ext_vector_type(4)));
typedef __attribute__((address_space(1))) v4i_* gb128_t;
typedef __attribute__((address_space(3))) v4i_* lb128_t;
typedef __attribute__((address_space(1))) int*  gb32_t;
typedef __attribute__((address_space(3))) int*  lb32_t;
// Route through integers: generic->int is always legal, int->as-ptr is always
// legal. Low 32 bits of a generic LDS address are the LDS byte offset (ISA 10.2).
#define GASYNC128(g, l) __builtin_amdgcn_global_load_async_to_lds_b128(          \
    (gb128_t)(unsigned long long)(g),                                            \
    (lb128_t)(unsigned)(unsigned long long)(l), 0, 0)
#define GASYNC32(g, l) __builtin_amdgcn_global_load_async_to_lds_b32(            \
    (gb32_t)(unsigned long long)(g),                                             \
    (lb32_t)(unsigned)(unsigned long long)(l), 0, 0)
#else
#define HAS_ASYNC_LDS 0
#endif

// ---------------------------------------------------------------- decode ----
__global__ __launch_bounds__(256) void decode_kernel(
    const float* __restrict__ rpn_class,
    const float4* __restrict__ rpn_bbox,
    const float4* __restrict__ anchors,
    float4* __restrict__ boxes,
    unsigned long long* __restrict__ keys) {
  int i = blockIdx.x * 256 + threadIdx.x;
  if (i >= N_ANCH) return;
  float s  = rpn_class[i];
  float4 d = rpn_bbox[i];
  float4 a = anchors[i];
  float d0 = s * d.x, d1 = s * d.y, d2 = s * d.z, d3 = s * d.w;
  float wdt = a.z - a.x, hgt = a.w - a.y;
  float cx = a.x + 0.5f * wdt + d0 * wdt;
  float cy = a.y + 0.5f * hgt + d1 * hgt;
  float nw = wdt * expf(d2);
  float nh = hgt * expf(d3);
  float4 b;
  b.x = fmaxf(cx - 0.5f * nw, 0.0f);
  b.y = fmaxf(cy - 0.5f * nh, 0.0f);
  b.z = fminf(cx + 0.5f * nw, 1023.0f);   // W-1
  b.w = fminf(cy + 0.5f * nh, 1023.0f);   // H-1
  boxes[i] = b;
  // Stable descending sort key: monotone-mapped score (inverted) | index.
  unsigned u = __float_as_uint(s);
  u = (u & 0x80000000u) ? ~u : (u | 0x80000000u);
  keys[i] = ((unsigned long long)(~u) << 32) | (unsigned)i;
}

// ---------------------------------------------------------------- sort ------
__device__ __forceinline__ void cmpex(unsigned long long& a, unsigned long long& b, bool asc) {
  unsigned long long lo = (a < b) ? a : b;
  unsigned long long hi = (a < b) ? b : a;
  a = asc ? lo : hi;
  b = asc ? hi : lo;
}

// One 1024-thread workgroup; 16384 u64 keys live in 128KB dynamic LDS
// (WGP has 320KB). Strides 1..8 run in registers (16 keys/thread),
// strides >=16 through LDS: 55 LDS passes instead of 105.
__global__ __launch_bounds__(1024) void sort_kernel(
    const unsigned long long* __restrict__ keysIn,
    const float4* __restrict__ boxes,
    float4* __restrict__ bs,
    float* __restrict__ area) {
  extern __shared__ unsigned long long sk[];   // 16384 entries
  const int t = threadIdx.x;
  unsigned long long r[16];
#pragma unroll
  for (int e = 0; e < 16; ++e) r[e] = keysIn[(t << 4) + e];
  // stages k = 1..4 entirely in registers (per-element direction)
#pragma unroll
  for (int k = 1; k <= 4; ++k) {
#pragma unroll
    for (int j = k - 1; j >= 0; --j) {
#pragma unroll
      for (int e = 0; e < 16; ++e) {
        if (!(e & (1 << j))) {
          bool asc = ((((t << 4) | e) >> k) & 1) == 0;
          cmpex(r[e], r[e | (1 << j)], asc);
        }
      }
    }
  }
#pragma unroll
  for (int e = 0; e < 16; ++e) sk[(t << 4) + e] = r[e];
  __syncthreads();

  for (int k = 5; k <= 14; ++k) {
    for (int j = k - 1; j >= 4; --j) {
#pragma unroll
      for (int p = 0; p < 8; ++p) {
        unsigned pi = (unsigned)(p * 1024 + t);
        unsigned i  = ((pi >> j) << (j + 1)) | (pi & ((1u << j) - 1u));
        unsigned ip = i | (1u << j);
        bool asc = ((i >> k) & 1u) == 0u;
        unsigned long long a = sk[i], b = sk[ip];
        if ((a > b) == asc) { sk[i] = b; sk[ip] = a; }
      }
      __syncthreads();
    }
    // register tail: strides 8,4,2,1 (direction uniform per thread for k>=5)
#pragma unroll
    for (int e = 0; e < 16; ++e) r[e] = sk[(t << 4) + e];
    bool asc = ((((unsigned)(t << 4)) >> k) & 1u) == 0u;
#pragma unroll
    for (int j = 3; j >= 0; --j) {
#pragma unroll
      for (int e = 0; e < 16; ++e)
        if (!(e & (1 << j))) cmpex(r[e], r[e | (1 << j)], asc);
    }
#pragma unroll
    for (int e = 0; e < 16; ++e) sk[(t << 4) + e] = r[e];
    __syncthreads();
  }
  // gather boxes into sorted order + precompute areas
  for (int i = t; i < N_ANCH; i += 1024) {
    unsigned idx = (unsigned)(sk[i] & 0xFFFFFFFFull);
    float4 b = boxes[idx];
    bs[i] = b;
    area[i] = (b.z - b.x) * (b.w - b.y);
  }
}

// ---------------------------------------------------------------- mask ------
// Each wave computes a 16-row x 32-col tile of the suppression bitmask.
// WMMA f32 16x16x4 produces S[r][c] = area_r + area_c per 16x16 half-tile:
//   A (16x4): col0 = area_row, col1 = 1   (VGPR0=K0 lanes0-15, VGPR1=K1; K2/K3 = 0)
//   B (4x16): row0 = 1, row1 = area_col   (VGPR0=K0 lanes0-15, VGPR1=K1; K2/K3 = 0)
// D layout (ISA 7.12.2): lane = column, VGPR v = row v (lanes 0-15) / v+8 (16-31),
// so a __ballot per VGPR yields two 16-bit row half-words directly.
// Tile boxes are staged global->LDS with GLOBAL_LOAD_ASYNC_TO_LDS (ASYNCcnt)
// when the toolchain exposes the builtins.
// Mask stored column-major: mask2[word * N + row] (coalesced store + scan reads).
__global__ __launch_bounds__(256) void iou_mask_kernel(
    const float4* __restrict__ bs,
    const float* __restrict__ area,
    unsigned* __restrict__ mask2) {
  const int w    = blockIdx.x;          // 32-column word index
  const int rowG = blockIdx.y << 7;     // 128 rows per block (8 waves x 16)
  const int wEnd = (w << 5) + 31;
  if (rowG > wEnd) return;              // strictly-lower-triangle words never read

  __shared__ float4 rbox[128];
  __shared__ float  ras[128];
  __shared__ float4 cbox[32];
  __shared__ float  cas[32];
  const int t = threadIdx.x;
#if HAS_ASYNC_LDS
  if (t < 128) {
    GASYNC128(bs + rowG + t,   &rbox[t]);
    GASYNC32 (area + rowG + t, &ras[t]);
  } else if (t < 160) {
    const int c = t - 128;
    GASYNC128(bs + (w << 5) + c,   &cbox[c]);
    GASYNC32 (area + (w << 5) + c, &cas[c]);
  }
  __builtin_amdgcn_s_wait_asynccnt(0);
#else
  if (t < 128) {
    rbox[t] = bs[rowG + t];
    ras[t]  = area[rowG + t];
  } else if (t < 160) {
    const int c = t - 128;
    cbox[c] = bs[(w << 5) + c];
    cas[c]  = area[(w << 5) + c];
  }
#endif
  __syncthreads();

  const int wave = t >> 5;
  const int lane = t & 31;
  const int rowBase = wave << 4;
  const int rAbs = rowG + rowBase;
  if (rAbs > wEnd) return;              // per-wave uniform skip

  const bool lo  = lane < 16;
  const int  l15 = lane & 15;

  v2f Amat; Amat[0] = lo ? ras[rowBase + l15] : 0.0f; Amat[1] = lo ? 1.0f : 0.0f;
  v2f Bm0;  Bm0[0]  = lo ? 1.0f : 0.0f;               Bm0[1]  = lo ? cas[l15]      : 0.0f;
  v2f Bm1;  Bm1[0]  = lo ? 1.0f : 0.0f;               Bm1[1]  = lo ? cas[16 + l15] : 0.0f;
  v8f Z = {0.f, 0.f, 0.f, 0.f, 0.f, 0.f, 0.f, 0.f};
  v8f S0 = __builtin_amdgcn_wmma_f32_16x16x4_f32(false, Amat, false, Bm0, (short)0, Z, false, false);
  v8f S1 = __builtin_amdgcn_wmma_f32_16x16x4_f32(false, Amat, false, Bm1, (short)0, Z, false, false);

  const float4 c0 = cbox[l15];
  const float4 c1 = cbox[16 + l15];
  const int rSel = rowBase + ((lane >> 4) << 3);  // rows v (lanes<16) / v+8 (lanes>=16)

  unsigned myword = 0;
#pragma unroll
  for (int v = 0; v < 8; ++v) {
    const float4 rb = rbox[rSel + v];
    // half 0 (columns w*32 .. +15)
    float ix1 = fmaxf(rb.x, c0.x), iy1 = fmaxf(rb.y, c0.y);
    float ix2 = fminf(rb.z, c0.z), iy2 = fminf(rb.w, c0.w);
    float inter = fmaxf(ix2 - ix1, 0.0f) * fmaxf(iy2 - iy1, 0.0f);
    float uni = fmaxf(S0[v] - inter, 1e-8f);
    unsigned ball0 = (unsigned)__ballot(inter > NMS_THR * uni);
    // half 1 (columns w*32+16 .. +31)
    ix1 = fmaxf(rb.x, c1.x); iy1 = fmaxf(rb.y, c1.y);
    ix2 = fminf(rb.z, c1.z); iy2 = fminf(rb.w, c1.w);
    inter = fmaxf(ix2 - ix1, 0.0f) * fmaxf(iy2 - iy1, 0.0f);
    uni = fmaxf(S1[v] - inter, 1e-8f);
    unsigned ball1 = (unsigned)__ballot(inter > NMS_THR * uni);

    unsigned wlow  = (ball0 & 0xFFFFu) | (ball1 << 16);        // tile row v
    unsigned whigh = (ball0 >> 16) | (ball1 & 0xFFFF0000u);    // tile row v+8
    myword = (lane == v)     ? wlow  : myword;
    myword = (lane == v + 8) ? whigh : myword;
  }
  if (lane < 16)
    mask2[(size_t)w * N_ANCH + (size_t)(rAbs + lane)] = myword;
}

// ---------------------------------------------------------------- scan ------
// Sequential greedy NMS over 512 word-blocks. Thread t owns removed-word t in
// a register. Per block: prefetch 32 mask words (8x uint4, contiguous thanks to
// column-major layout) before the barrier; owner thread resolves its 32 boxes
// from LDS-cached diagonal words; one barrier; everyone ORs kept rows.
__global__ __launch_bounds__(512) void nms_scan_kernel(
    const unsigned* __restrict__ mask2,
    const float4* __restrict__ bs,
    float* __restrict__ out) {
  extern __shared__ unsigned sm[];
  unsigned* diag  = sm;            // 16384
  unsigned* keepW = sm + N_ANCH;   // 512
  unsigned* base  = keepW + 512;   // 513
  const int t = threadIdx.x;

  for (int rr = t; rr < N_ANCH; rr += 512)
    diag[rr] = mask2[((size_t)(rr >> 5)) * N_ANCH + (size_t)rr];
  __syncthreads();

  unsigned remv = 0;
  const uint4* colBase = (const uint4*)(mask2 + (size_t)t * N_ANCH);
  for (int i = 0; i < 512; ++i) {
    uint4 m[8];
    const bool active = (t > i);
    if (active) {
#pragma unroll
      for (int q = 0; q < 8; ++q) m[q] = colBase[(i << 3) + q];
    }
    if (t == i) {
      unsigned sup = remv;
      const int rb = i << 5;
#pragma unroll
      for (int b = 0; b < 32; ++b) {
        unsigned mm = diag[rb + b];                 // unconditional: hoistable
        if (!((sup >> b) & 1u)) sup |= mm & (0xFFFFFFFEu << b);
      }
      keepW[i] = ~sup;
    }
    __syncthreads();
    const unsigned kw = keepW[i];
    if (active) {
      unsigned acc = 0;
#pragma unroll
      for (int q = 0; q < 8; ++q) {
        const int bb = q << 2;
        acc |= m[q].x & (unsigned)(-(int)((kw >> (bb + 0)) & 1u));
        acc |= m[q].y & (unsigned)(-(int)((kw >> (bb + 1)) & 1u));
        acc |= m[q].z & (unsigned)(-(int)((kw >> (bb + 2)) & 1u));
        acc |= m[q].w & (unsigned)(-(int)((kw >> (bb + 3)) & 1u));
      }
      remv |= acc;
    }
  }
  __syncthreads();
  if (t == 0) {
    unsigned acc = 0;
    for (int i = 0; i < 512; ++i) { base[i] = acc; acc += __popc(keepW[i]); }
    base[512] = acc;
  }
  __syncthreads();
  const unsigned total = base[512];
  unsigned kw = keepW[t];
  unsigned rk = base[t];
  while (kw) {
    int b = __ffs(kw) - 1;
    kw &= kw - 1u;
    if (rk < PROPS) {
      float4 bx = bs[(t << 5) + b];
      out[rk * 4 + 0] = bx.x; out[rk * 4 + 1] = bx.y;
      out[rk * 4 + 2] = bx.z; out[rk * 4 + 3] = bx.w;
    }
    ++rk;
  }
  for (unsigned rz = total + (unsigned)t; rz < PROPS; rz += 512) {
    out[rz * 4 + 0] = 0.f; out[rz * 4 + 1] = 0.f;
    out[rz * 4 + 2] = 0.f; out[rz * 4 + 3] = 0.f;
  }
}

// ---------------------------------------------------------------- launch ----
extern "C" void kernel_launch(void* const* d_in, const int* in_sizes, int n_in,
                              void* d_out, int out_size, void* d_ws, size_t ws_size,
                              hipStream_t stream) {
  (void)in_sizes; (void)n_in; (void)out_size; (void)ws_size;
  const float*  rpn_class = (const float*)d_in[0];
  const float4* rpn_bbox  = (const float4*)d_in[1];
  const float4* anchors   = (const float4*)d_in[2];
  // d_in[3] (image) only contributes H/W = 1024 constants.

  char* ws = (char*)d_ws;
  float4*             boxes = (float4*)(ws);                        // 256 KB
  float4*             bs    = (float4*)(ws + (256u << 10));         // 256 KB
  float*              area  = (float*)(ws + (512u << 10));          //  64 KB
  unsigned long long* keys  = (unsigned long long*)(ws + (576u << 10)); // 128 KB
  unsigned*           mask2 = (unsigned*)(ws + (704u << 10));       //  32 MB

  decode_kernel<<<N_ANCH / 256, 256, 0, stream>>>(rpn_class, rpn_bbox, anchors, boxes, keys);
  sort_kernel<<<1, 1024, (size_t)N_ANCH * sizeof(unsigned long long), stream>>>(keys, boxes, bs, area);
  iou_mask_kernel<<<dim3(NWORDS, N_ANCH / 128), 256, 0, stream>>>(bs, area, mask2);
  nms_scan_kernel<<<1, 512, (size_t)(N_ANCH + 512 + 513) * sizeof(unsigned), stream>>>(mask2, bs, (float*)d_out);
}